// MultiHeadAttention_23416161698246
// MI455X (gfx1250) — compile-verified
//
#include <hip/hip_runtime.h>
#include <hip/hip_bf16.h>

// ---------------------------------------------------------------------------
// Performer/FAVOR+ attention, fp32, on MI455X (gfx1250).
//
// Head-broadcast structure in the reference makes qp/kp/den head-independent,
// so the whole pipeline reduces to dense per-batch GEMMs (~8.6 GFLOP fp32)
// against ~tens of MB of HBM traffic (23.3 TB/s => ~2us). Compute-bound in
// the fp32 matrix pipe => every GEMM uses V_WMMA_F32_16X16X4_F32 (fp32 in,
// fp32 acc: the only WMMA that preserves the reference's fp32 numerics;
// bf16 inputs would give ~1e-2 rel. error after exp()). Epilogues (bias,
// RoPE, exp/rowmax, 1/den) are fused into the GEMM kernels. No float
// atomics anywhere => deterministic.
// ---------------------------------------------------------------------------

typedef __attribute__((ext_vector_type(2))) float v2f;
typedef __attribute__((ext_vector_type(8))) float v8f;

// D = A(16x4,f32) * B(4x16,f32) + C(16x16,f32)
__device__ __forceinline__ v8f wmma4(v2f a, v2f b, v8f c) {
  return __builtin_amdgcn_wmma_f32_16x16x4_f32(
      /*neg_a=*/false, a, /*neg_b=*/false, b,
      /*c_mod=*/(short)0, c, /*reuse_a=*/false, /*reuse_b=*/false);
}

// Fixed problem sizes
constexpr int cB = 4, cL = 2048, cD = 256, cM = 256;
constexpr long cS = (long)cL * cD;  // per-batch activation stride (== L*M)

// ---------------------------------------------------------------------------
// Generic 64x64-tile GEMM, 256 threads = 8 waves (wave32), each wave owns a
// 32x16 strip (two 16x16 accumulators, B-fragment reused). K staged in LDS
// in kc=16 panels. All leading dims are 256. TA: A is stored K-major
// (used for KVS = KP^T @ V).
// EPI: 0 = (+bias if non-null)            -> plain / V-proj / num / KVS
//      1 = +bias, then RoPE * 0.25        -> Q/K proj (needs pos)
//      2 = *(1/rowdiv) + bias             -> output GEMM (needs den)
// ---------------------------------------------------------------------------
template <int EPI, bool TA>
__global__ __launch_bounds__(256) void gemm64(
    const float* __restrict__ A, long sA,
    const float* __restrict__ Bm, long sB,
    const float* __restrict__ bias,
    const float* __restrict__ pos,
    const float* __restrict__ rowdiv,
    float* __restrict__ C, long sC, int K) {
  const int bz = blockIdx.z;
  const int row0 = blockIdx.x * 64;
  const int col0 = blockIdx.y * 64;
  A += (long)bz * sA;
  Bm += (long)bz * sB;
  C += (long)bz * sC;

  __shared__ float As[64][17];
  __shared__ float Bs[16][68];
  __shared__ float Cs[(EPI == 1) ? 64 : 1][65];  // only for RoPE pair mixing

  const int t = threadIdx.x;
  const int lane = t & 31;
  const int w = t >> 5;
  const int mr0 = (w >> 2) * 32;  // wave's row strip base (0 or 32)
  const int nc = (w & 3) * 16;    // wave's 16-col tile base

  v8f acc0 = {};
  v8f acc1 = {};

  for (int ko = 0; ko < K; ko += 16) {
    if constexpr (TA) {  // A global is [K][256-rows]: As[m][k] = A[ko+k][row0+m]
      const int k = t >> 4, m4 = (t & 15) * 4;
      const float4 x = *(const float4*)&A[(long)(ko + k) * 256 + row0 + m4];
      As[m4 + 0][k] = x.x; As[m4 + 1][k] = x.y;
      As[m4 + 2][k] = x.z; As[m4 + 3][k] = x.w;
    } else {             // A global row-major [rows][256]
      const int r = t >> 2, c4 = (t & 3) * 4;
      const float4 x = *(const float4*)&A[(long)(row0 + r) * 256 + ko + c4];
      As[r][c4 + 0] = x.x; As[r][c4 + 1] = x.y;
      As[r][c4 + 2] = x.z; As[r][c4 + 3] = x.w;
    }
    {  // B global [K][256] row-major
      const int k = t >> 4, n4 = (t & 15) * 4;
      const float4 x = *(const float4*)&Bm[(long)(ko + k) * 256 + col0 + n4];
      Bs[k][n4 + 0] = x.x; Bs[k][n4 + 1] = x.y;
      Bs[k][n4 + 2] = x.z; Bs[k][n4 + 3] = x.w;
    }
    __syncthreads();
#pragma unroll
    for (int kk = 0; kk < 16; kk += 4) {
      // 16x16x4 f32 fragments: lane row m = lane&15, k0 = kk + (lane>=16)*2
      const int k0 = kk + ((lane >> 4) << 1);
      const int ml = lane & 15;
      v2f a0, a1, bf;
      a0.x = As[mr0 + ml][k0];      a0.y = As[mr0 + ml][k0 + 1];
      a1.x = As[mr0 + 16 + ml][k0]; a1.y = As[mr0 + 16 + ml][k0 + 1];
      bf.x = Bs[k0][nc + ml];       bf.y = Bs[k0 + 1][nc + ml];
      acc0 = wmma4(a0, bf, acc0);
      acc1 = wmma4(a1, bf, acc1);
    }
    __syncthreads();
  }

  // C layout: VGPR g -> row mr + g + 8*(lane>=16), col nc + (lane&15)
  const int colL = nc + (lane & 15);
  const int rb = (lane >> 4) << 3;

  if constexpr (EPI == 1) {  // bias then RoPE(pos)*0.25 (pairs mix columns -> LDS)
    const float bb = bias[col0 + colL];
#pragma unroll
    for (int g = 0; g < 8; ++g) {
      Cs[mr0 + rb + g][colL] = acc0[g] + bb;
      Cs[mr0 + 16 + rb + g][colL] = acc1[g] + bb;
    }
    __syncthreads();
    const int r = t >> 2;           // 64 rows, 4 threads/row
    const int p0 = (t & 3) * 16;    // 16 cols = 8 even/odd pairs per thread
    const long l = row0 + r;        // row within batch (pos index)
#pragma unroll
    for (int pp = 0; pp < 16; pp += 2) {
      const int cg = col0 + p0 + pp;
      const float e = Cs[r][p0 + pp];
      const float o = Cs[r][p0 + pp + 1];
      const float cs = pos[l * 256 + cg + 1];  // cos at odd slots
      const float sn = pos[l * 256 + cg];      // sin at even slots
      C[l * 256 + cg]     = 0.25f * (e * cs - o * sn);  // 256^-0.25 folded in
      C[l * 256 + cg + 1] = 0.25f * (o * cs + e * sn);
    }
  } else {
    const float bb = bias ? bias[col0 + colL] : 0.0f;
#pragma unroll
    for (int g = 0; g < 8; ++g) {
      const int r0g = mr0 + rb + g;
      const int r1g = mr0 + 16 + rb + g;
      float v0 = acc0[g], v1 = acc1[g];
      if constexpr (EPI == 2) {
        v0 /= rowdiv[(long)bz * cL + row0 + r0g];
        v1 /= rowdiv[(long)bz * cL + row0 + r1g];
      }
      C[(long)(row0 + r0g) * 256 + col0 + colL] = v0 + bb;
      C[(long)(row0 + r1g) * 256 + col0 + colL] = v1 + bb;
    }
  }
}

// ---------------------------------------------------------------------------
// Feature-map GEMM (NT): xp[l,m] = x[l,:] . proj[m,:], tile = 32 rows x all
// 256 cols so each workgroup owns complete rows (needed for the row max).
// EPI 0: qp = (1/16)*(exp(xp - diag - rowmax(xp)) + 1e-6)   (query side)
// EPI 1: write raw xp, emit per-workgroup max (key side, global max later)
// ---------------------------------------------------------------------------
template <int EPI>
__global__ __launch_bounds__(256) void gemm_nt_feat(
    const float* __restrict__ X, const float* __restrict__ P,
    const float* __restrict__ dg, float* __restrict__ out,
    float* __restrict__ wgmax) {
  (void)dg; (void)wgmax;
  const int b = blockIdx.z;
  const int row0 = blockIdx.x * 32;
  X += (long)b * cS;
  out += (long)b * cS;

  __shared__ float As[32][17];
  __shared__ float Bs[16][257];
  __shared__ float Cs[32][257];
  __shared__ float red[32][9];
  __shared__ float rowm[32];

  const int t = threadIdx.x;
  const int lane = t & 31;
  const int w = t >> 5;
  const int mr0 = (w >> 2) * 16;  // wave row tile (0 or 16)
  const int ncb = (w & 3) * 64;   // wave col strip (4 x 16-wide tiles)

  v8f z = {};
  v8f acc[4] = {z, z, z, z};

  for (int ko = 0; ko < 256; ko += 16) {
    {  // A: 32x16 panel of x
      const int r = t >> 3, c2 = (t & 7) * 2;
      const float2 x = *(const float2*)&X[(long)(row0 + r) * 256 + ko + c2];
      As[r][c2] = x.x; As[r][c2 + 1] = x.y;
    }
    {  // B^T: Bs[k][n] = proj[n][ko+k]; thread t loads proj row t's segment
      const int n = t;
#pragma unroll
      for (int q4 = 0; q4 < 16; q4 += 4) {
        const float4 x = *(const float4*)&P[(long)n * 256 + ko + q4];
        Bs[q4 + 0][n] = x.x; Bs[q4 + 1][n] = x.y;
        Bs[q4 + 2][n] = x.z; Bs[q4 + 3][n] = x.w;
      }
    }
    __syncthreads();
#pragma unroll
    for (int kk = 0; kk < 16; kk += 4) {
      const int k0 = kk + ((lane >> 4) << 1);
      const int ml = lane & 15;
      v2f a; a.x = As[mr0 + ml][k0]; a.y = As[mr0 + ml][k0 + 1];
#pragma unroll
      for (int j = 0; j < 4; ++j) {
        v2f bf;
        bf.x = Bs[k0][ncb + 16 * j + ml];
        bf.y = Bs[k0 + 1][ncb + 16 * j + ml];
        acc[j] = wmma4(a, bf, acc[j]);
      }
    }
    __syncthreads();
  }

  {  // stage full 32x256 tile for row reductions
    const int rb = (lane >> 4) << 3;
    const int cl = lane & 15;
#pragma unroll
    for (int j = 0; j < 4; ++j)
#pragma unroll
      for (int g = 0; g < 8; ++g)
        Cs[mr0 + rb + g][ncb + 16 * j + cl] = acc[j][g];
  }
  __syncthreads();

  const int r = t >> 3, s8 = t & 7;  // 8 threads per row
  float pm = -3.402823466e38f;
  for (int c = s8; c < 256; c += 8) pm = fmaxf(pm, Cs[r][c]);
  red[r][s8] = pm;
  __syncthreads();
  if (t < 32) {
    float m = red[t][0];
#pragma unroll
    for (int i = 1; i < 8; ++i) m = fmaxf(m, red[t][i]);
    if constexpr (EPI == 0)
      rowm[t] = m + dg[(long)b * cL + row0 + t];  // fold diag into the shift
    else
      rowm[t] = m;
  }
  __syncthreads();

  if constexpr (EPI == 0) {
    for (int c = s8; c < 256; c += 8) {
      const float v = expf(Cs[r][c] - rowm[r]);
      out[(long)(row0 + r) * 256 + c] = 0.0625f * (v + 1e-6f);  // M^-0.5=1/16
    }
  } else {
    for (int c = s8; c < 256; c += 8)
      out[(long)(row0 + r) * 256 + c] = Cs[r][c];
    if (t == 0) {
      float m = rowm[0];
#pragma unroll
      for (int i = 1; i < 32; ++i) m = fmaxf(m, rowm[i]);
      wgmax[b * 64 + blockIdx.x] = m;
    }
  }
}

// ---------------------------------------------------------------------------
// Small deterministic helper kernels
// ---------------------------------------------------------------------------

// diag[b,l] = 0.5 * sum_d x[b,l,d]^2   (one wave per row, 8 rows per block)
__global__ __launch_bounds__(256) void row_halfsumsq(
    const float* __restrict__ X, float* __restrict__ dg) {
  const int row = blockIdx.x * 8 + (threadIdx.x >> 5);
  const int lane = threadIdx.x & 31;
  const float* x = X + (long)row * 256;
  float s = 0.0f;
  for (int c = lane; c < 256; c += 32) { const float v = x[c]; s += v * v; }
  for (int m = 16; m; m >>= 1) s += __shfl_xor(s, m, 32);
  if (lane == 0) dg[row] = 0.5f * s;
}

// kmax[b] = max over 64 workgroup maxima
__global__ __launch_bounds__(64) void reduce_kmax(
    const float* __restrict__ wgmax, float* __restrict__ kmax) {
  __shared__ float s[64];
  const int b = blockIdx.x, t = threadIdx.x;
  s[t] = wgmax[b * 64 + t];
  __syncthreads();
  for (int st = 32; st; st >>= 1) {
    if (t < st) s[t] = fmaxf(s[t], s[t + st]);
    __syncthreads();
  }
  if (t == 0) kmax[b] = s[0];
}

// kp = (1/16)*(exp(xp - diag[b,l] - kmax[b]) + 1e-6), in place
__global__ __launch_bounds__(256) void kp_exp(
    float* __restrict__ xp, const float* __restrict__ dg,
    const float* __restrict__ kmax) {
  const long i = (long)blockIdx.x * 256 + threadIdx.x;
  const int bl = (int)(i >> 8);   // row index b*L + l   (M = 256)
  const int b = bl >> 11;         // L = 2048
  xp[i] = 0.0625f * (expf(xp[i] - dg[bl] - kmax[b]) + 1e-6f);
}

// ksum[b,m] = sum_l kp[b,l,m] in two deterministic stages
__global__ __launch_bounds__(256) void ksum_part(
    const float* __restrict__ kp, float* __restrict__ part) {
  const int chunk = blockIdx.x, b = blockIdx.y, m = threadIdx.x;
  const float* p = kp + (long)b * cS + (long)chunk * 128 * 256 + m;
  float s = 0.0f;
#pragma unroll 4
  for (int l = 0; l < 128; ++l) s += p[(long)l * 256];
  part[((long)b * 16 + chunk) * 256 + m] = s;
}
__global__ __launch_bounds__(256) void ksum_final(
    const float* __restrict__ part, float* __restrict__ ksum) {
  const int b = blockIdx.x, m = threadIdx.x;
  float s = 0.0f;
#pragma unroll
  for (int c = 0; c < 16; ++c) s += part[((long)b * 16 + c) * 256 + m];
  ksum[b * 256 + m] = s;
}

// den[b,l] = dot(qp[b,l,:], ksum[b,:])  (one wave per row)
__global__ __launch_bounds__(256) void den_kernel(
    const float* __restrict__ qp, const float* __restrict__ ksum,
    float* __restrict__ den) {
  const int row = blockIdx.x * 8 + (threadIdx.x >> 5);
  const int lane = threadIdx.x & 31;
  const int b = row >> 11;  // L = 2048
  const float* q = qp + (long)row * 256;
  const float* ks = ksum + b * 256;
  float s = 0.0f;
  for (int c = lane; c < 256; c += 32) s += q[c] * ks[c];
  for (int m = 16; m; m >>= 1) s += __shfl_xor(s, m, 32);
  if (lane == 0) den[row] = s;
}

// ---------------------------------------------------------------------------
// Orchestration
// ---------------------------------------------------------------------------
extern "C" void kernel_launch(void* const* d_in, const int* in_sizes, int n_in,
                              void* d_out, int out_size, void* d_ws,
                              size_t ws_size, hipStream_t stream) {
  (void)in_sizes; (void)n_in; (void)out_size; (void)ws_size;
  const float* query = (const float*)d_in[0];
  const float* key   = (const float*)d_in[1];
  const float* value = (const float*)d_in[2];
  // d_in[3] mask: unused by the reference
  const float* Wq = (const float*)d_in[4];  const float* bq = (const float*)d_in[5];
  const float* Wk = (const float*)d_in[6];  const float* bk = (const float*)d_in[7];
  const float* Wv = (const float*)d_in[8];  const float* bv = (const float*)d_in[9];
  const float* Wo = (const float*)d_in[10]; const float* bo = (const float*)d_in[11];
  const float* proj = (const float*)d_in[12];
  const float* pos  = (const float*)d_in[13];
  float* out = (float*)d_out;

  // Workspace carve-up (floats); ~51.5 MB total, every block written before read.
  const long BLD = (long)cB * cL * cD;  // 2,097,152
  float* p = (float*)d_ws;
  float* xq = p;  p += BLD;             // 0.25*rope(q)   [B,L,D]
  float* xk = p;  p += BLD;             // 0.25*rope(k)
  float* vb = p;  p += BLD;             // v projection
  float* qp = p;  p += BLD;             // query features [B,L,M]
  float* kp = p;  p += BLD;             // raw xp_k, then kp in place
  float* nm = p;  p += BLD;             // num = qp @ kvs
  float* kvs = p; p += (long)cB * cM * cD;
  float* kspart = p; p += (long)cB * 16 * cM;
  float* ksum = p; p += (long)cB * cM;
  float* dq = p;  p += (long)cB * cL;
  float* dk = p;  p += (long)cB * cL;
  float* den = p; p += (long)cB * cL;
  float* wgm = p; p += (long)cB * 64;
  float* kmx = p; p += cB;

  const dim3 blk(256);
  const dim3 gProj(cL / 64, cD / 64, cB);  // (32,4,4)

  // 1) projections (+bias), RoPE*scale fused for q/k
  gemm64<1, false><<<gProj, blk, 0, stream>>>(query, cS, Wq, 0, bq, pos, nullptr, xq, cS, cD);
  gemm64<1, false><<<gProj, blk, 0, stream>>>(key,   cS, Wk, 0, bk, pos, nullptr, xk, cS, cD);
  gemm64<0, false><<<gProj, blk, 0, stream>>>(value, cS, Wv, 0, bv, nullptr, nullptr, vb, cS, cD);

  // 2) diag terms
  row_halfsumsq<<<dim3(cB * cL / 8), blk, 0, stream>>>(xq, dq);
  row_halfsumsq<<<dim3(cB * cL / 8), blk, 0, stream>>>(xk, dk);

  // 3) feature maps (x @ proj^T with fused epilogues)
  gemm_nt_feat<0><<<dim3(cL / 32, 1, cB), blk, 0, stream>>>(xq, proj, dq, qp, nullptr);
  gemm_nt_feat<1><<<dim3(cL / 32, 1, cB), blk, 0, stream>>>(xk, proj, nullptr, kp, wgm);

  // 4) key-side global max, then kp = exp(...)
  reduce_kmax<<<dim3(cB), dim3(64), 0, stream>>>(wgm, kmx);
  kp_exp<<<dim3((unsigned)(BLD / 256)), blk, 0, stream>>>(kp, dk, kmx);

  // 5) ksum (two-stage deterministic column sum)
  ksum_part<<<dim3(16, cB), blk, 0, stream>>>(kp, kspart);
  ksum_final<<<dim3(cB), blk, 0, stream>>>(kspart, ksum);

  // 6) KVS = KP^T @ V   (TN, K = L = 2048)
  gemm64<0, true><<<dim3(cM / 64, cD / 64, cB), blk, 0, stream>>>(
      kp, cS, vb, cS, nullptr, nullptr, nullptr, kvs, (long)cM * cD, cL);

  // 7) den = qp . ksum
  den_kernel<<<dim3(cB * cL / 8), blk, 0, stream>>>(qp, ksum, den);

  // 8) num = qp @ kvs
  gemm64<0, false><<<gProj, blk, 0, stream>>>(qp, cS, kvs, (long)cM * cD,
                                              nullptr, nullptr, nullptr, nm, cS, cM);

  // 9) out = (num/den) @ Wo + bo
  gemm64<2, false><<<gProj, blk, 0, stream>>>(nm, cS, Wo, 0, bo, nullptr, den,
                                              out, cS, cD);
}